// GNNArma_71141838291378
// MI455X (gfx1250) — compile-verified
//
#include <hip/hip_runtime.h>
#include <hip/hip_bf16.h>

#define NTOT   91136
#define NGRAPH 32
#define NNODE  2848
#define EDGES  1000000
#define KST    3
#define TLAY   5
#define HID    32

typedef __attribute__((ext_vector_type(2))) float v2f;
typedef __attribute__((ext_vector_type(8))) float v8f;

// ---------------------------------------------------------------- utilities
__global__ void zero_kernel(float* __restrict__ p, long n) {
  long i = (long)blockIdx.x * blockDim.x + threadIdx.x;
  long stride = (long)gridDim.x * blockDim.x;
  for (; i < n; i += stride) p[i] = 0.f;
}

__global__ void relu_kernel(float* __restrict__ p, long n) {
  long i = (long)blockIdx.x * blockDim.x + threadIdx.x;
  long stride = (long)gridDim.x * blockDim.x;
  for (; i < n; i += stride) p[i] = fmaxf(p[i], 0.f);
}

// ---------------------------------------------------------------- gcn_norm
__global__ void deg_kernel(const long long* __restrict__ ei, const float* __restrict__ w,
                           float* __restrict__ deg) {
  int e = blockIdx.x * blockDim.x + threadIdx.x;
  if (e >= EDGES) return;
  atomicAdd(&deg[(int)ei[EDGES + e]], w[e]);
}

__global__ void norm_kernel(const long long* __restrict__ ei, const float* __restrict__ w,
                            const float* __restrict__ deg, float* __restrict__ norm) {
  int e = blockIdx.x * blockDim.x + threadIdx.x;
  if (e >= EDGES) return;
  int r = (int)ei[e];
  int c = (int)ei[EDGES + e];
  float dr = deg[r], dc = deg[c];
  float a = dr > 0.f ? rsqrtf(dr) : 0.f;
  float b = dc > 0.f ? rsqrtf(dc) : 0.f;
  norm[e] = a * w[e] * b;
}

// ---------------------------------------------------------------- WMMA GEMM
// C[k][m][o] = sum_f A(k)[m][f] * W(k)[f][o] (+ bias(k)[o])
// V_WMMA_F32_16X16X4_F32; FIN/FOUT are compile-time so all padding masks fold
// to branchless v_cndmask on clamped in-bounds loads (no exec churn, no
// per-element branches). Each wave: preload B frags once, sweep MT row tiles.
// A frag (16x4 f32): lanes 0-15 -> M=lane, K={0,1}; lanes 16-31 -> M=l, K={2,3}.
// B frag (4x16 f32): lanes 0-15 -> N=lane, K={0,1}; lanes 16-31 -> N=l, K={2,3}.
// C/D (16x16 f32):   lane<16 -> N=lane, M=r;       lanes 16-31 -> N=l, M=8+r.
template<int FIN, int FOUT, int MT>
__global__ void mm_wmma(const float* __restrict__ A, long aK,
                        const float* __restrict__ W, long wK,
                        const float* __restrict__ bias, long bK,
                        float* __restrict__ C, long cK) {
  constexpr int STEPS = (FIN + 3) / 4;
  constexpr int COLT  = (FOUT + 15) / 16;
  int rowGroups = (NTOT / 16) / MT;               // 5696 / MT, exact
  int perK = rowGroups * COLT;
  int wid = blockIdx.x * (blockDim.x >> 5) + (threadIdx.x >> 5);
  if (wid >= KST * perK) return;                  // wave-uniform exit
  int k  = wid / perK;
  int rm = wid - k * perK;
  int mg = rm / COLT;
  int nt = rm - mg * COLT;

  int lane = threadIdx.x & 31;
  int lh   = lane >> 4;                           // half-wave id
  int l    = lane & 15;
  int nCol = (nt << 4) + l;                       // < FOUT when FOUT%16==0

  const float* Ak = A + (long)k * aK;
  const float* Wk = W + (long)k * wK;

  // ---- preload B fragments (reused across MT row tiles)
  v2f bf[STEPS];
#pragma unroll
  for (int s = 0; s < STEPS; ++s) {
    int k0 = 4 * s + 2 * lh;
    if constexpr (FOUT >= 16) {
      if constexpr (FIN % 4 == 0) {
        bf[s].x = Wk[(long)k0 * FOUT + nCol];
        bf[s].y = Wk[(long)(k0 + 1) * FOUT + nCol];
      } else {                                    // FIN == 2: rows 2,3 are padding
        float r0 = Wk[nCol];
        float r1 = Wk[FOUT + nCol];
        bf[s].x = (lh == 0) ? r0 : 0.f;
        bf[s].y = (lh == 0) ? r1 : 0.f;
      }
    } else {                                      // FOUT == 1: only column 0 real
      if constexpr (FIN % 4 == 0) {
        float r0 = Wk[k0];
        float r1 = Wk[k0 + 1];
        bf[s].x = (nCol == 0) ? r0 : 0.f;
        bf[s].y = (nCol == 0) ? r1 : 0.f;
      } else {                                    // FIN == 1, FOUT == 1
        float r0 = Wk[0];
        bf[s].x = (nCol == 0 && lh == 0) ? r0 : 0.f;
        bf[s].y = 0.f;
      }
    }
  }

  float bv = 0.f;
  if (bias != nullptr) {
    if constexpr (FOUT >= 16) bv = bias[(long)k * bK + nCol];
    else                      bv = (nCol == 0) ? bias[(long)k * bK] : 0.f;
  }

  // ---- sweep MT row tiles with the cached B fragments
  for (int mt = 0; mt < MT; ++mt) {
    int mBase = (mg * MT + mt) << 4;
    int mA = mBase + l;
    v8f acc = {0.f, 0.f, 0.f, 0.f, 0.f, 0.f, 0.f, 0.f};
#pragma unroll
    for (int s = 0; s < STEPS; ++s) {
      v2f af;
      if constexpr (FIN % 4 == 0) {               // aligned b64 load, no masks
        af = *(const v2f*)(Ak + (long)mA * FIN + 4 * s + 2 * lh);
      } else if constexpr (FIN == 2) {            // K={2,3} half is padding
        v2f t = *(const v2f*)(Ak + (long)mA * 2);
        af.x = (lh == 0) ? t.x : 0.f;
        af.y = (lh == 0) ? t.y : 0.f;
      } else {                                    // FIN == 1
        float t = Ak[mA];
        af.x = (lh == 0) ? t : 0.f;
        af.y = 0.f;
      }
      acc = __builtin_amdgcn_wmma_f32_16x16x4_f32(false, af, false, bf[s],
                                                  (short)0, acc, false, false);
    }
    if constexpr (FOUT >= 16) {
      float* Ck = C + (long)k * cK + (long)mBase * FOUT + nCol;
#pragma unroll
      for (int r = 0; r < 8; ++r)
        Ck[(long)((lh << 3) + r) * FOUT] = acc[r] + bv;
    } else {                                      // FOUT == 1
      if (nCol == 0) {
        float* Ck = C + (long)k * cK + mBase;
#pragma unroll
        for (int r = 0; r < 8; ++r)
          Ck[(lh << 3) + r] = acc[r] + bv;
      }
    }
  }
}

// ---------------------------------------------------------------- edge scatter
// agg[k][col[e]][o] += h[k][row[e]][o] * norm[e]
__global__ void scatter_kernel(const long long* __restrict__ ei, const float* __restrict__ norm,
                               const float* __restrict__ h, float* __restrict__ agg,
                               int Fout, long knF) {
  long tid = (long)blockIdx.x * blockDim.x + threadIdx.x;
  long total = (long)EDGES * Fout;
  if (tid >= total) return;
  int e = (int)(tid / Fout);
  int o = (int)(tid - (long)e * Fout);
  int r = (int)ei[e];
  int c = (int)ei[EDGES + e];
  float nv = norm[e];
#pragma unroll
  for (int k = 0; k < KST; ++k) {
    float v = h[(long)k * knF + (long)r * Fout + o] * nv;
    atomicAdd(&agg[(long)k * knF + (long)c * Fout + o], v);
  }
}

// ---------------------------------------------------------------- mean over K
__global__ void meank_kernel(const float* __restrict__ b0, float* __restrict__ y,
                             long nf, long knF) {
  long i = (long)blockIdx.x * blockDim.x + threadIdx.x;
  long stride = (long)gridDim.x * blockDim.x;
  for (; i < nf; i += stride)
    y[i] = (b0[i] + b0[i + knF] + b0[i + 2 * knF]) * (1.f / 3.f);
}

// ---------------------------------------------------------------- batch norm
// Requires (gridDim*blockDim) % F == 0 so each thread owns one channel.
__global__ void bn_stats(const float* __restrict__ x, float* __restrict__ sums,
                         long total, int F) {
  __shared__ float s[64];
  if (threadIdx.x < 2 * F) s[threadIdx.x] = 0.f;
  __syncthreads();
  long i0 = (long)blockIdx.x * blockDim.x + threadIdx.x;
  long stride = (long)gridDim.x * blockDim.x;
  if (i0 < total) {
    int o = (int)(i0 % F);
    float ls = 0.f, lq = 0.f;
    for (long i = i0; i < total; i += stride) { float v = x[i]; ls += v; lq += v * v; }
    atomicAdd(&s[o], ls);
    atomicAdd(&s[F + o], lq);
  }
  __syncthreads();
  if (threadIdx.x < 2 * F) atomicAdd(&sums[threadIdx.x], s[threadIdx.x]);
}

__global__ void bn_finalize(const float* __restrict__ sums, const float* __restrict__ g,
                            const float* __restrict__ be, float* __restrict__ ss, int F) {
  int o = threadIdx.x;
  if (o >= F) return;
  float inv = 1.f / (float)NTOT;
  float mean = sums[o] * inv;
  float var  = sums[F + o] * inv - mean * mean;
  float sc = g[o] * rsqrtf(var + 1e-5f);
  ss[o] = sc;
  ss[F + o] = be[o] - mean * sc;
}

__global__ void bn_apply(float* __restrict__ x, const float* __restrict__ ss,
                         long total, int F) {
  long i = (long)blockIdx.x * blockDim.x + threadIdx.x;
  long stride = (long)gridDim.x * blockDim.x;
  for (; i < total; i += stride) {
    int o = (int)(i % F);
    x[i] = fmaxf(x[i] * ss[o] + ss[F + o], 0.f);
  }
}

// ---------------------------------------------------------------- final head
__global__ void final_kernel(const float* __restrict__ h, const float* __restrict__ cw,
                             const float* __restrict__ cb, float* __restrict__ out) {
  __shared__ float sm[256];
  int g = blockIdx.x;
  float wv = cw[0], bv = cb[0];
  float mx = -3.402823466e38f;
  for (int i = threadIdx.x; i < NNODE; i += blockDim.x) {
    int n = g * NNODE + i;
    float v = h[n] * wv + bv;
    out[n] = v;
    mx = fmaxf(mx, v);
  }
  sm[threadIdx.x] = mx;
  __syncthreads();
  for (int s = 128; s > 0; s >>= 1) {
    if (threadIdx.x < s) sm[threadIdx.x] = fmaxf(sm[threadIdx.x], sm[threadIdx.x + s]);
    __syncthreads();
  }
  if (threadIdx.x == 0) out[NTOT + g] = sm[0];
}

// ---------------------------------------------------------------- host side
template<int FIN, int FOUT>
static void launch_mm(hipStream_t s, const float* A, long aK, const float* W, long wK,
                      const float* bias, long bK, float* C, long cK) {
  constexpr int MT = 4;                               // 5696 % 4 == 0
  constexpr int COLT = (FOUT + 15) / 16;
  int rowGroups = (NTOT / 16) / MT;
  int waves = KST * rowGroups * COLT;
  int blocks = (waves + 3) / 4;                       // 4 waves (128 thr) per block
  mm_wmma<FIN, FOUT, MT><<<blocks, 128, 0, s>>>(A, aK, W, wK, bias, bK, C, cK);
}

template<int FIN, int FOUT>
static void arma_layer(hipStream_t s, const float* xin,
                       const float* iw, const float* w, const float* rw, const float* b,
                       const long long* ei, const float* norm,
                       float* buf0 /*out/agg*/, float* buf1 /*h*/) {
  long knF = (long)NTOT * FOUT;
  for (int t = 0; t < TLAY; ++t) {
    // h = (t==0 ? x @ iw : out @ w[t-1])  -> buf1
    if (t == 0)
      launch_mm<FIN, FOUT>(s, xin, 0, iw, (long)FIN * FOUT, nullptr, 0, buf1, knF);
    else
      launch_mm<FOUT, FOUT>(s, buf0, knF, w + (long)(t - 1) * KST * FOUT * FOUT,
                            (long)FOUT * FOUT, nullptr, 0, buf1, knF);
    // agg = x @ rw[t] + b[t]  -> buf0 (root term initializes the accumulator)
    launch_mm<FIN, FOUT>(s, xin, 0, rw + (long)t * KST * FIN * FOUT,
                         (long)FIN * FOUT, b + (long)t * KST * FOUT, FOUT, buf0, knF);
    // agg += scatter(norm * h[row]) at col
    long totalS = (long)EDGES * FOUT;
    scatter_kernel<<<(int)((totalS + 255) / 256), 256, 0, s>>>(ei, norm, buf1, buf0, FOUT, knF);
    // out = relu(agg) in place
    relu_kernel<<<4096, 256, 0, s>>>(buf0, KST * knF);
  }
}

static void post_layer(hipStream_t s, const float* buf0, int Fout,
                       const float* g, const float* be,
                       float* xbuf, float* sums, float* ss) {
  long nf = (long)NTOT * Fout;
  meank_kernel<<<4096, 256, 0, s>>>(buf0, xbuf, nf, nf);
  zero_kernel<<<1, 64, 0, s>>>(sums, 64);
  bn_stats<<<512, 256, 0, s>>>(xbuf, sums, nf, Fout);   // 512*256 divisible by 32 and 1
  bn_finalize<<<1, 32, 0, s>>>(sums, g, be, ss, Fout);
  bn_apply<<<4096, 256, 0, s>>>(xbuf, ss, nf, Fout);
}

extern "C" void kernel_launch(void* const* d_in, const int* in_sizes, int n_in,
                              void* d_out, int out_size, void* d_ws, size_t ws_size,
                              hipStream_t stream) {
  (void)in_sizes; (void)n_in; (void)out_size; (void)ws_size;

  const float*     x   = (const float*)d_in[0];        // [N,2]
  const long long* ei  = (const long long*)d_in[1];    // [2,E] int64
  const float*     ew  = (const float*)d_in[2];        // [E]
  // d_in[3] = batch (layout is known: NGRAPH x NNODE contiguous) -> unused
  const float* iw1 = (const float*)d_in[4];
  const float* w1  = (const float*)d_in[5];
  const float* rw1 = (const float*)d_in[6];
  const float* b1  = (const float*)d_in[7];
  const float* g1  = (const float*)d_in[8];
  const float* be1 = (const float*)d_in[9];
  const float* iw2 = (const float*)d_in[10];
  const float* w2  = (const float*)d_in[11];
  const float* rw2 = (const float*)d_in[12];
  const float* b2  = (const float*)d_in[13];
  const float* g2  = (const float*)d_in[14];
  const float* be2 = (const float*)d_in[15];
  const float* iw3 = (const float*)d_in[16];
  const float* w3  = (const float*)d_in[17];
  const float* rw3 = (const float*)d_in[18];
  const float* b3  = (const float*)d_in[19];
  const float* g3  = (const float*)d_in[20];
  const float* be3 = (const float*)d_in[21];
  const float* cw  = (const float*)d_in[22];
  const float* cb  = (const float*)d_in[23];
  float* out = (float*)d_out;

  // workspace layout (floats)
  float* ws   = (float*)d_ws;
  float* deg  = ws;                                    // N
  float* norm = deg + NTOT;                            // E
  float* buf0 = norm + EDGES;                          // K*N*32
  float* buf1 = buf0 + (long)KST * NTOT * HID;         // K*N*32
  float* xbuf = buf1 + (long)KST * NTOT * HID;         // N*32
  float* sums = xbuf + (long)NTOT * HID;               // 64
  float* ss   = sums + 64;                             // 64

  // gcn_norm
  zero_kernel<<<1024, 256, 0, stream>>>(deg, NTOT);
  deg_kernel<<<(EDGES + 255) / 256, 256, 0, stream>>>(ei, ew, deg);
  norm_kernel<<<(EDGES + 255) / 256, 256, 0, stream>>>(ei, ew, deg, norm);

  // ARMAConv 1: 2 -> 32, BN + ReLU
  arma_layer<2, HID>(stream, x, iw1, w1, rw1, b1, ei, norm, buf0, buf1);
  post_layer(stream, buf0, HID, g1, be1, xbuf, sums, ss);

  // ARMAConv 2: 32 -> 32, BN + ReLU
  arma_layer<HID, HID>(stream, xbuf, iw2, w2, rw2, b2, ei, norm, buf0, buf1);
  post_layer(stream, buf0, HID, g2, be2, xbuf, sums, ss);

  // ARMAConv 3: 32 -> 1, BN + ReLU
  arma_layer<HID, 1>(stream, xbuf, iw3, w3, rw3, b3, ei, norm, buf0, buf1);
  post_layer(stream, buf0, 1, g3, be3, xbuf, sums, ss);

  // classifier + per-graph max: d_out = [logits_nodes (N)] ++ [logits_graph (32)]
  final_kernel<<<NGRAPH, 256, 0, stream>>>(xbuf, cw, cb, out);
}